// GCN_61314953118523
// MI455X (gfx1250) — compile-verified
//
#include <hip/hip_runtime.h>

typedef __attribute__((ext_vector_type(2))) float v2f;
typedef __attribute__((ext_vector_type(8))) float v8f;

#define NNODES 100000
#define NEDGES 3200000
#define FIN    128
#define HID    16

// ---------------------------------------------------------------------------
// Degree / normalization
// ---------------------------------------------------------------------------
__global__ void init_deg_kernel(float* __restrict__ deg, int n) {
    int i = blockIdx.x * blockDim.x + threadIdx.x;
    if (i < n) deg[i] = 1.0f;  // self-loop contribution
}

__global__ void deg_count_kernel(const long long* __restrict__ dst,
                                 float* __restrict__ deg, int e) {
    int i = blockIdx.x * blockDim.x + threadIdx.x;
    if (i < e) unsafeAtomicAdd(&deg[(int)dst[i]], 1.0f);
}

__global__ void dinv_kernel(float* __restrict__ deg, int n) {
    int i = blockIdx.x * blockDim.x + threadIdx.x;
    if (i < n) deg[i] = rsqrtf(deg[i]);   // deg >= 1 always (self-loops)
}

// ---------------------------------------------------------------------------
// hs = (A @ W) * dinv[row]   (A: n x K, W: K x 16) via V_WMMA_F32_16X16X4_F32.
// One wave computes one 16x16 output tile. Stores the PRE-SCALED features
// hs[i] = h[i]*dinv[i] (so the edge scatter only needs dinv[dst]) and seeds
// the scatter accumulator with the self-loop message agg = hs[i]*dinv[i]
// ( == h[i]*dinv[i]^2 ).
//
// A-frag (16x4): lane L -> row L&15, K pair = k0 + 2*(L>>4)
// B-frag (4x16): lane L -> col L&15, K pair = k0 + 2*(L>>4)
// C/D (16x16 f32, 8 VGPR): vgpr v -> row v + 8*(L>>4), col L&15
// ---------------------------------------------------------------------------
__global__ void gcn_gemm_wmma(const float* __restrict__ A,
                              const float* __restrict__ W,
                              const float* __restrict__ dinv,
                              float* __restrict__ hs,
                              float* __restrict__ agg,
                              int nTiles, int K) {
    int wave = threadIdx.x >> 5;
    int lane = threadIdx.x & 31;
    int tile = blockIdx.x * (blockDim.x >> 5) + wave;
    if (tile >= nTiles) return;            // wave-uniform: EXEC stays all-ones
    int m0  = tile << 4;
    int col = lane & 15;                   // A-row for lane == B-col for lane
    int hi  = lane >> 4;

    const float* arow = A + (long)(m0 + col) * K;

    v8f c = {};
    for (int k0 = 0; k0 < K; k0 += 4) {
        int kb = k0 + (hi << 1);
        v2f a, b;
        a.x = arow[kb];
        a.y = arow[kb + 1];
        b.x = W[kb * 16 + col];
        b.y = W[(kb + 1) * 16 + col];
        c = __builtin_amdgcn_wmma_f32_16x16x4_f32(
            /*neg_a=*/false, a, /*neg_b=*/false, b,
            /*c_mod=*/(short)0, c, /*reuse_a=*/false, /*reuse_b=*/false);
    }

#pragma unroll
    for (int v = 0; v < 8; ++v) {
        int   m      = m0 + v + (hi << 3);
        float dv     = dinv[m];
        float scaled = c[v] * dv;               // h * dinv[i]
        hs[m * 16 + col]  = scaled;
        agg[m * 16 + col] = scaled * dv;        // self-loop: h * dinv[i]^2
    }
}

// ---------------------------------------------------------------------------
// Edge scatter: agg[dst] += hs[src] * dinv[dst]   (hs already has dinv[src])
// One thread per edge; float4 gathers of the 64B pre-scaled source row;
// 16 fire-and-forget f32 atomics (global_atomic_add_f32, non-returning) into
// the L2-resident accumulator.
// ---------------------------------------------------------------------------
__global__ void scatter_kernel(const long long* __restrict__ src,
                               const long long* __restrict__ dst,
                               const float* __restrict__ dinv,
                               const float* __restrict__ hs,
                               float* __restrict__ agg, int e) {
    int i = blockIdx.x * blockDim.x + threadIdx.x;
    if (i >= e) return;
    int s = (int)src[i];
    int d = (int)dst[i];
    float w = dinv[d];
    const float4* hp = (const float4*)(hs + (long)s * 16);
    float* ad = agg + (long)d * 16;
#pragma unroll
    for (int q = 0; q < 4; ++q) {
        float4 m = hp[q];
        unsafeAtomicAdd(ad + 4 * q + 0, m.x * w);
        unsafeAtomicAdd(ad + 4 * q + 1, m.y * w);
        unsafeAtomicAdd(ad + 4 * q + 2, m.z * w);
        unsafeAtomicAdd(ad + 4 * q + 3, m.w * w);
    }
}

// ---------------------------------------------------------------------------
// Elementwise epilogues
// ---------------------------------------------------------------------------
__global__ void bias_relu_kernel(const float* __restrict__ agg,
                                 const float* __restrict__ b,
                                 float* __restrict__ act, int total) {
    int i = blockIdx.x * blockDim.x + threadIdx.x;
    if (i < total) {
        float v = agg[i] + b[i & 15];
        act[i] = v > 0.0f ? v : 0.0f;
    }
}

__global__ void bias_out_kernel(const float* __restrict__ agg,
                                const float* __restrict__ b,
                                float* __restrict__ out, int total) {
    int i = blockIdx.x * blockDim.x + threadIdx.x;
    if (i < total) out[i] = agg[i] + b[i & 15];
}

// ---------------------------------------------------------------------------
extern "C" void kernel_launch(void* const* d_in, const int* in_sizes, int n_in,
                              void* d_out, int out_size, void* d_ws, size_t ws_size,
                              hipStream_t stream) {
    const float*     x   = (const float*)d_in[0];
    const long long* ei  = (const long long*)d_in[1];  // int64 per reference
    const float*     W1  = (const float*)d_in[2];
    const float*     b1  = (const float*)d_in[3];
    const float*     W2  = (const float*)d_in[4];
    const float*     b2  = (const float*)d_in[5];
    float*           out = (float*)d_out;

    const long long* srcI = ei;            // edge_index[0, :]
    const long long* dstI = ei + NEDGES;   // edge_index[1, :]

    // Workspace layout (floats). Total = 100000 + 5 * 1.6M = 8.1M fl = 32.4 MB.
    float* ws   = (float*)d_ws;
    float* dinv = ws;                      // N        (deg, then dinv in-place)
    float* hs1  = ws + NNODES;             // N*16  pre-scaled layer-1 features
    float* agg1 = hs1 + (size_t)NNODES * 16;
    float* act  = agg1 + (size_t)NNODES * 16;
    float* hs2  = act + (size_t)NNODES * 16;
    float* agg2 = hs2 + (size_t)NNODES * 16;

    const int B = 256;
    const int nTiles = NNODES / 16;        // 6250, exact
    const int gemmGrid = (nTiles + 7) / 8; // 8 waves (wave32) per 256-thr block

    init_deg_kernel<<<(NNODES + B - 1) / B, B, 0, stream>>>(dinv, NNODES);
    deg_count_kernel<<<(NEDGES + B - 1) / B, B, 0, stream>>>(dstI, dinv, NEDGES);
    dinv_kernel<<<(NNODES + B - 1) / B, B, 0, stream>>>(dinv, NNODES);

    // Layer 1: transform (+dinv pre-scale, self-loop seed), scatter, bias+ReLU
    gcn_gemm_wmma<<<gemmGrid, B, 0, stream>>>(x, W1, dinv, hs1, agg1, nTiles, FIN);
    scatter_kernel<<<(NEDGES + B - 1) / B, B, 0, stream>>>(srcI, dstI, dinv, hs1, agg1, NEDGES);
    bias_relu_kernel<<<(NNODES * 16 + B - 1) / B, B, 0, stream>>>(agg1, b1, act, NNODES * 16);

    // Layer 2: transform (+pre-scale, self-loop seed), scatter, bias -> output
    gcn_gemm_wmma<<<gemmGrid, B, 0, stream>>>(act, W2, dinv, hs2, agg2, nTiles, HID);
    scatter_kernel<<<(NEDGES + B - 1) / B, B, 0, stream>>>(srcI, dstI, dinv, hs2, agg2, NEDGES);
    bias_out_kernel<<<(NNODES * 16 + B - 1) / B, B, 0, stream>>>(agg2, b2, out, NNODES * 16);
}